// UpdateBlock_13005160972653
// MI455X (gfx1250) — compile-verified
//
#include <hip/hip_runtime.h>

// ---------------------------------------------------------------------------
// Graph message-passing update for MI455X (gfx1250):
//   acc = segment_sum(scaled_v, nbrs[:,0])      [50000 x 256]
//   out = x_i + acc @ W^T + b                   [50000 x 128]
//
// Roofline: scaled_v = 1.024 GB streamed once -> ~47 us floor at 23.3 TB/s.
// Naive scatter needs 256M f32 atomics (L2 RMW-rate bound, ~10x the stream
// time). Instead: counting-sort edge IDs by destination (2M tiny u32 atomics
// on L2-resident counters), then GATHER per node with NT b128 loads and zero
// data atomics. GEMM runs on v_wmma_f32_16x16x4_f32 (3.3 GFLOP, negligible).
// ---------------------------------------------------------------------------

typedef __attribute__((ext_vector_type(2))) float v2f;
typedef __attribute__((ext_vector_type(4))) float v4f;
typedef __attribute__((ext_vector_type(8))) float v8f;

#define N_NODES 50000
#define FEAT    128              // output feature dim
#define HF      256              // num_heads * feat_dim (accumulator width)
#define N_EDGES 1000000
#define STRIPS  (N_NODES / 16)   // 3125 row-strips of 16

// Workspace layout (bytes):
#define OFS_COUNTS   (0)                     // u32[50000]  histogram
#define OFS_OFFSETS  (256 * 1024)            // u32[50000]  exclusive prefix (start)
#define OFS_CURSOR   (512 * 1024)            // u32[50000]  running fill cursor
#define OFS_EDGEIDS  (768 * 1024)            // u32[1M]     edge ids sorted by dst
#define OFS_ACC      (5 * 1024 * 1024)       // f32[50000*256] accumulator
// total = 5 MB + 51.2 MB = 56.4 MB

// ---- K1: zero the histogram ----------------------------------------------
__global__ void zero_counts_kernel(unsigned* __restrict__ counts) {
    int i = blockIdx.x * blockDim.x + threadIdx.x;
    if (i < N_NODES) counts[i] = 0u;
}

// ---- K2: histogram of destination nodes ----------------------------------
__global__ void __launch_bounds__(256)
hist_kernel(const int* __restrict__ nbrs, unsigned* __restrict__ counts) {
    int e = blockIdx.x * blockDim.x + threadIdx.x;
    if (e >= N_EDGES) return;
    int dst = nbrs[(long)e * 2];
    atomicAdd(&counts[dst], 1u);
}

// ---- K3: exclusive prefix sum (single block, Hillis-Steele in LDS) -------
__global__ void __launch_bounds__(1024)
scan_kernel(const unsigned* __restrict__ counts,
            unsigned* __restrict__ offsets,
            unsigned* __restrict__ cursor) {
    __shared__ unsigned buf[1024];
    __shared__ unsigned carry;
    if (threadIdx.x == 0) carry = 0u;
    __syncthreads();

    for (int base = 0; base < N_NODES; base += 1024) {
        const unsigned c = carry;                   // stable since last barrier
        const int i = base + (int)threadIdx.x;
        unsigned v = (i < N_NODES) ? counts[i] : 0u;
        buf[threadIdx.x] = v;
        __syncthreads();
#pragma unroll
        for (int d = 1; d < 1024; d <<= 1) {
            unsigned t = (threadIdx.x >= (unsigned)d) ? buf[threadIdx.x - d] : 0u;
            __syncthreads();
            buf[threadIdx.x] += t;
            __syncthreads();
        }
        const unsigned excl = buf[threadIdx.x] - v + c;
        if (i < N_NODES) { offsets[i] = excl; cursor[i] = excl; }
        __syncthreads();
        if (threadIdx.x == 0) carry = c + buf[1023];
        __syncthreads();
    }
}

// ---- K4: bucket-fill edge IDs by destination -----------------------------
__global__ void __launch_bounds__(256)
fill_kernel(const int* __restrict__ nbrs,
            unsigned* __restrict__ cursor,
            unsigned* __restrict__ edge_ids) {
    int e = blockIdx.x * blockDim.x + threadIdx.x;
    if (e >= N_EDGES) return;
    int dst = nbrs[(long)e * 2];
    unsigned pos = atomicAdd(&cursor[dst], 1u);
    edge_ids[pos] = (unsigned)e;
}

// ---- K5: gather-accumulate, one wave per node ----------------------------
// 32 lanes x 8 floats cover the 256-wide row entirely in registers.
// scaled_v rows are read exactly once -> non-temporal; acc written once (RT,
// stays in L2 for the GEMM).
__global__ void __launch_bounds__(128)
gather_kernel(const unsigned* __restrict__ offsets,
              const unsigned* __restrict__ counts,
              const unsigned* __restrict__ edge_ids,
              const float* __restrict__ sv,
              float* __restrict__ acc) {
    const int wave = threadIdx.x >> 5;
    const int lane = threadIdx.x & 31;
    const int node = blockIdx.x * 4 + wave;       // uniform per wave
    if (node >= N_NODES) return;                  // whole-wave exit

    const unsigned start = offsets[node];
    const unsigned end   = start + counts[node];

    v4f s0 = {0.f, 0.f, 0.f, 0.f};
    v4f s1 = {0.f, 0.f, 0.f, 0.f};

    unsigned e = start;
    for (; e + 2 <= end; e += 2) {                // 2-way unroll for MLP
        const unsigned e0 = edge_ids[e];
        const unsigned e1 = edge_ids[e + 1];
        const v4f* p0 = (const v4f*)(sv + (long)e0 * HF + lane * 8);
        const v4f* p1 = (const v4f*)(sv + (long)e1 * HF + lane * 8);
        v4f a0 = __builtin_nontemporal_load(p0);
        v4f b0 = __builtin_nontemporal_load(p0 + 1);
        v4f a1 = __builtin_nontemporal_load(p1);
        v4f b1 = __builtin_nontemporal_load(p1 + 1);
        s0 += a0 + a1;
        s1 += b0 + b1;
    }
    if (e < end) {
        const unsigned e0 = edge_ids[e];
        const v4f* p0 = (const v4f*)(sv + (long)e0 * HF + lane * 8);
        s0 += __builtin_nontemporal_load(p0);
        s1 += __builtin_nontemporal_load(p0 + 1);
    }

    v4f* dst = (v4f*)(acc + (long)node * HF + lane * 8);
    dst[0] = s0;
    dst[1] = s1;
}

// ---- K6: out = x_i + acc @ W^T + b  via V_WMMA_F32_16X16X4_F32 -----------
// One wave per 16-row strip; wave computes all 8 column tiles so each acc
// row is loaded exactly once. ISA layouts per CDNA5 ISA:
//   A 16x4 f32 : lane m = lane&15 holds K = k+2*(lane>>4), +1
//   B 4x16 f32 : lane n = lane&15 holds same K pair, column n
//   C/D 16x16  : VGPR i -> M = i + 8*(lane>>4), N = lane&15
__global__ void __launch_bounds__(128)
gemm_wmma_kernel(const float* __restrict__ acc,
                 const float* __restrict__ W,     // [128, 256] row-major
                 const float* __restrict__ xi,    // [50000, 128]
                 const float* __restrict__ bias,  // [128]
                 float* __restrict__ out) {       // [50000, 128]
    const int wave  = threadIdx.x >> 5;
    const int lane  = threadIdx.x & 31;
    const int strip = blockIdx.x * 4 + wave;      // uniform per wave
    if (strip >= STRIPS) return;                  // whole-wave exit: EXEC all-1

    const int half = lane >> 4;
    const int nrow = lane & 15;

    v8f c[8];
#pragma unroll
    for (int nt = 0; nt < 8; ++nt)
        c[nt] = (v8f){0.f, 0.f, 0.f, 0.f, 0.f, 0.f, 0.f, 0.f};

    const float* arow = acc + (long)strip * 16 * HF + (long)nrow * HF;

    for (int k = 0; k < HF; k += 4) {
        const int kk = k + 2 * half;
        v2f a = *(const v2f*)(arow + kk);
#pragma unroll
        for (int nt = 0; nt < 8; ++nt) {
            const int f = nt * 16 + nrow;
            v2f b = *(const v2f*)(W + (long)f * HF + kk);
            c[nt] = __builtin_amdgcn_wmma_f32_16x16x4_f32(
                        false, a, false, b, (short)0, c[nt], false, false);
        }
    }

#pragma unroll
    for (int nt = 0; nt < 8; ++nt) {
        const int   f  = nt * 16 + nrow;
        const float bb = bias[f];
#pragma unroll
        for (int i = 0; i < 8; ++i) {
            const long m = (long)strip * 16 + i + 8 * half;
            out[m * FEAT + f] = xi[m * FEAT + f] + c[nt][i] + bb;
        }
    }
}

// ---------------------------------------------------------------------------
extern "C" void kernel_launch(void* const* d_in, const int* in_sizes, int n_in,
                              void* d_out, int out_size, void* d_ws, size_t ws_size,
                              hipStream_t stream) {
    const int*   nbrs = (const int*)d_in[0];     // [1M, 2] int
    const float* xi   = (const float*)d_in[1];   // [50000, 128]
    const float* sv   = (const float*)d_in[2];   // [1M, 256]
    const float* W    = (const float*)d_in[3];   // [128, 256]
    const float* bias = (const float*)d_in[4];   // [128]
    float*       out  = (float*)d_out;           // [50000, 128]

    char* ws = (char*)d_ws;
    unsigned* counts   = (unsigned*)(ws + OFS_COUNTS);
    unsigned* offsets  = (unsigned*)(ws + OFS_OFFSETS);
    unsigned* cursor   = (unsigned*)(ws + OFS_CURSOR);
    unsigned* edge_ids = (unsigned*)(ws + OFS_EDGEIDS);
    float*    acc      = (float*)(ws + OFS_ACC);

    (void)in_sizes; (void)n_in; (void)out_size; (void)ws_size;

    // 1) zero histogram (everything else is fully overwritten downstream)
    zero_counts_kernel<<<(N_NODES + 255) / 256, 256, 0, stream>>>(counts);

    // 2) histogram destinations (1M u32 atomics on 200 KB, L2-resident)
    hist_kernel<<<(N_EDGES + 255) / 256, 256, 0, stream>>>(nbrs, counts);

    // 3) exclusive scan -> bucket start offsets + fill cursors
    scan_kernel<<<1, 1024, 0, stream>>>(counts, offsets, cursor);

    // 4) place edge IDs into per-destination buckets
    fill_kernel<<<(N_EDGES + 255) / 256, 256, 0, stream>>>(nbrs, cursor, edge_ids);

    // 5) gather-accumulate per node (zero data atomics, NT streaming reads)
    gather_kernel<<<(N_NODES + 3) / 4, 128, 0, stream>>>(offsets, counts, edge_ids,
                                                         sv, acc);

    // 6) WMMA GEMM + residual + bias
    gemm_wmma_kernel<<<(STRIPS + 3) / 4, 128, 0, stream>>>(acc, W, xi, bias, out);
}